// ScaledDotProductAttention_17489106829910
// MI455X (gfx1250) — compile-verified
//
#include <hip/hip_runtime.h>

typedef __attribute__((ext_vector_type(2))) float v2f;
typedef __attribute__((ext_vector_type(8))) float v8f;

#define N_     325
#define DK_    32
#define NT_    21          // ceil(325/16) tiles
#define PCOLS  352         // padded LDS row length
#define HEADS  768         // B*T*H = 8*12*8

__device__ __forceinline__ int imin(int a, int b) { return a < b ? a : b; }

__global__ __launch_bounds__(32)
void edge_attn_kernel(const float* __restrict__ Q,
                      const float* __restrict__ K,
                      const float* __restrict__ V,
                      const float* __restrict__ E,
                      float* __restrict__ outCtx,
                      float* __restrict__ outE)
{
    __shared__ float P[16 * PCOLS];   // unnormalized exp(e - max) rows
    __shared__ float rsum[16];

    const int lane = threadIdx.x;          // 0..31 (wave32)
    const int m    = lane & 15;            // row/col within 16-half
    const int h    = (lane >> 4) << 1;     // K-offset 0 or 2 (A/B frag layout)
    const int hi8  = (lane >> 4) << 3;     // C/D row offset 0 or 8

    const int head  = blockIdx.y;
    const int qbase = blockIdx.x * 16;
    const bool fullQ = (qbase + 16) <= N_; // uniform: all 16 q-rows valid

    const float* Qh = Q + (size_t)head * (N_ * DK_);
    const float* Kh = K + (size_t)head * (N_ * DK_);
    const float* Vh = V + (size_t)head * (N_ * DK_);
    const float* Eh = E + (size_t)head * ((size_t)N_ * N_);
    float* ctxh = outCtx + (size_t)head * (N_ * DK_);
    float* eh   = outE   + (size_t)head * ((size_t)N_ * N_);

    const float scale = 0.17677669529663687f;  // 1/sqrt(32)

    // ---- Q A-fragments: clamped rows, unconditional b64 loads ----
    v2f aq[8];
    {
        const float* Qr = Qh + imin(qbase + m, N_ - 1) * DK_;
        #pragma unroll
        for (int c = 0; c < 8; ++c)
            aq[c] = *(const v2f*)(Qr + 4 * c + h);
    }

    // ---- S = scale * Q K^T ; e = S * E : emit e_, stage rows in LDS ----
    for (int j = 0; j < NT_; ++j) {
        const float* Kr = Kh + imin(j * 16 + m, N_ - 1) * DK_;
        v8f acc = {};
        #pragma unroll
        for (int c = 0; c < 8; ++c) {
            v2f b = *(const v2f*)(Kr + 4 * c + h);
            acc = __builtin_amdgcn_wmma_f32_16x16x4_f32(
                      false, aq[c], false, b, (short)0, acc, false, false);
        }
        const int col = j * 16 + m;

        if (fullQ & (j < 20)) {
            // -------- fast path: every lane/row/col valid, no predication ----
            const float* Ep = Eh + (size_t)(qbase + hi8) * N_ + col;
            float*       ep = eh + (size_t)(qbase + hi8) * N_ + col;
            float eld[8];
            #pragma unroll
            for (int r = 0; r < 8; ++r)           // clause-batched NT loads
                eld[r] = __builtin_nontemporal_load(Ep + r * N_);
            #pragma unroll
            for (int r = 0; r < 8; ++r) {
                const float ev = acc[r] * scale * eld[r];
                __builtin_nontemporal_store(ev, ep + r * N_);
                P[(r + hi8) * PCOLS + col] = ev;
            }
        } else {
            // -------- boundary path: guarded stores, clamped loads ----------
            const int ccl = imin(col, N_ - 1);
            #pragma unroll
            for (int r = 0; r < 8; ++r) {
                const int  rowl  = r + hi8;
                const int  row   = qbase + rowl;
                const bool valid = (row < N_) & (col < N_);
                const float eld = __builtin_nontemporal_load(
                                      Eh + imin(row, N_ - 1) * N_ + ccl);
                const float ev  = valid ? acc[r] * scale * eld
                                        : -__builtin_inff();
                if (valid)
                    __builtin_nontemporal_store(ev, eh + row * N_ + col);
                P[rowl * PCOLS + col] = ev;   // pad cols/rows hold -inf
            }
        }
    }
    __syncthreads();

    // ---- row softmax: max + exp + sum (wave32 shfl_xor reductions) ----
    #pragma unroll 1
    for (int r = 0; r < 16; ++r) {
        float mx = -__builtin_inff();
        for (int c = lane; c < 336; c += 32)
            mx = fmaxf(mx, P[r * PCOLS + c]);
        #pragma unroll
        for (int s = 16; s > 0; s >>= 1)
            mx = fmaxf(mx, __shfl_xor(mx, s, 32));

        float sum = 0.0f;
        for (int c = lane; c < 336; c += 32) {
            float x = __expf(P[r * PCOLS + c] - mx);
            P[r * PCOLS + c] = x;         // unnormalized prob
            sum += x;
        }
        #pragma unroll
        for (int s = 16; s > 0; s >>= 1)
            sum += __shfl_xor(sum, s, 32);
        if (lane == 0) rsum[r] = sum;
    }
    __syncthreads();

    // ---- ctx = P V : K-dim 325 (82 steps of 4), two 16x16 N-tiles ----
    // P cols >= 325 are exactly 0, so V row index may be clamped freely.
    v8f c0 = {}, c1 = {};
    #pragma unroll 4
    for (int kk = 0; kk < 82; ++kk) {
        const int k0 = kk * 4 + h;              // lane covers K rows k0, k0+1
        v2f a = *(const v2f*)(&P[m * PCOLS + k0]);   // ds_load_b64
        const float* Vr0 = Vh + imin(k0,     N_ - 1) * DK_;
        const float* Vr1 = Vh + imin(k0 + 1, N_ - 1) * DK_;
        v2f bx, by;
        bx.x = Vr0[m];      bx.y = Vr1[m];
        by.x = Vr0[m + 16]; by.y = Vr1[m + 16];
        c0 = __builtin_amdgcn_wmma_f32_16x16x4_f32(
                 false, a, false, bx, (short)0, c0, false, false);
        c1 = __builtin_amdgcn_wmma_f32_16x16x4_f32(
                 false, a, false, by, (short)0, c1, false, false);
    }

    #pragma unroll
    for (int r = 0; r < 8; ++r) {
        const int rowl = r + hi8;
        const int row  = qbase + rowl;
        if (row < N_) {
            const float inv = 1.0f / rsum[rowl];
            ctxh[row * DK_ + m]      = c0[r] * inv;
            ctxh[row * DK_ + m + 16] = c1[r] * inv;
        }
    }
}

extern "C" void kernel_launch(void* const* d_in, const int* in_sizes, int n_in,
                              void* d_out, int out_size, void* d_ws, size_t ws_size,
                              hipStream_t stream) {
    (void)in_sizes; (void)n_in; (void)out_size; (void)d_ws; (void)ws_size;
    const float* Q = (const float*)d_in[0];
    const float* K = (const float*)d_in[1];
    const float* V = (const float*)d_in[2];
    const float* E = (const float*)d_in[3];
    float* outCtx = (float*)d_out;                                   // [768,325,32]
    float* outE   = outCtx + (size_t)HEADS * N_ * DK_;               // [768,325,325]
    dim3 grid(NT_, HEADS);
    edge_attn_kernel<<<grid, 32, 0, stream>>>(Q, K, V, E, outCtx, outE);
}